// VectorQuantizer_24618752541167
// MI455X (gfx1250) — compile-verified
//
#include <hip/hip_runtime.h>
#include <hip/hip_bf16.h>

// ---------------------------------------------------------------------------
// VQ-VAE vector quantizer for MI455X (gfx1250).
//   inputs:  d_in[0] = inputs    [32, 256, 32, 32] f32   (NCHW)
//            d_in[1] = embedding [1024, 256]       f32
//   output:  d_out = quantized(8388608 f32) ++ loss(1 f32) ++ indices(32768 f32)
// Core GEMM (32768 x 1024 x 256) on exact-fp32 V_WMMA_F32_16X16X4_F32, with
// double-buffered GLOBAL_LOAD_ASYNC_TO_LDS staging of the codebook.
// ---------------------------------------------------------------------------

typedef float v2f __attribute__((ext_vector_type(2)));
typedef float v8f __attribute__((ext_vector_type(8)));

#define NROWS   32768      // 32 * 32 * 32 flattened vectors
#define DIM     256
#define KCODES  1024
#define QELEMS  8388608    // 32*256*32*32
#define LDS_PAD 4
#define LDS_W   (DIM + LDS_PAD)   // 260 floats per codebook row in LDS

#if defined(__gfx1250__) && __has_builtin(__builtin_amdgcn_global_load_async_to_lds_b128)
#define HAVE_ASYNC_LDS 1
// Builtin signature (from hipcc diagnostic): param0 = int4-vector AS1*, so:
typedef int b128_t __attribute__((vector_size(16)));
typedef b128_t __attribute__((address_space(1))) gb128_t;   // global int4*
typedef b128_t __attribute__((address_space(3))) lb128_t;   // LDS int4*
#else
#define HAVE_ASYNC_LDS 0
#endif

static __device__ __forceinline__ void wait_async_zero() {
#if HAVE_ASYNC_LDS
#if __has_builtin(__builtin_amdgcn_s_wait_asynccnt)
  __builtin_amdgcn_s_wait_asynccnt(0);
#else
  asm volatile("s_wait_asynccnt 0x0" ::: "memory");
#endif
#endif
}

// ---------------------------------------------------------------------------
// Kernel 1: eTe[k] = sum_d embedding[k][d]^2
// ---------------------------------------------------------------------------
__global__ void k_ete(const float* __restrict__ emb, float* __restrict__ eTe) {
  int k = blockIdx.x * 256 + threadIdx.x;             // 4 blocks x 256 = 1024
  const float4* row = (const float4*)(emb + (size_t)k * DIM);
  float s = 0.f;
#pragma unroll 8
  for (int i = 0; i < DIM / 4; ++i) {
    float4 v = row[i];
    s += v.x * v.x + v.y * v.y + v.z * v.z + v.w * v.w;
  }
  eTe[k] = s;
}

// ---------------------------------------------------------------------------
// Kernel 2: WMMA distance GEMM + running argmin.
//   block = 256 threads (8 waves). Wave w owns rows [blk*128 + w*16, +16).
//   32 tile-pairs of 32 codebook rows, double-buffered in LDS via async DMA.
// ---------------------------------------------------------------------------
__global__ void __launch_bounds__(256)
k_argmin(const float* __restrict__ in,     // NCHW inputs
         const float* __restrict__ emb,    // [K][D]
         const float* __restrict__ eTe,    // [K]
         int*   __restrict__ idxbuf,       // [NROWS] int result
         float* __restrict__ idx_out_f) {  // [NROWS] float copy into d_out
  __shared__ float eL[2][32 * LDS_W];      // 2 x 33,280 B

  const int tid  = threadIdx.x;
  const int lane = tid & 31;
  const int wave = tid >> 5;
  const int nn   = lane & 15;              // column-slot / row-in-tile id
  const int c0   = (lane >> 4) << 1;       // K-pair base: 0 (lanes 0-15) or 2

  const int mbase = blockIdx.x * 128 + wave * 16;
  const int row   = mbase + nn;            // this lane's A-matrix row (M = nn)
  const int bimg  = row >> 10;             // image index (1024 rows per image)
  const int hw    = row & 1023;
  const float* xbase = in + (size_t)bimg * (DIM * 1024) + hw;  // NCHW: +c*1024

  // Staging lambda: copy 32 codebook rows [kbase, kbase+32) into eL[buf].
  auto stage = [&](int buf, int kbase) {
#pragma unroll
    for (int i = tid; i < 32 * (DIM / 4); i += 256) {
      int r  = i >> 6;                     // row 0..31
      int dc = (i & 63) << 2;              // d offset (float4 granularity)
      const float* g = emb + (size_t)(kbase + r) * DIM + dc;
      float*       l = &eL[buf][r * LDS_W + dc];   // (r*260+dc)*4B is 16B aligned
#if HAVE_ASYNC_LDS
      __builtin_amdgcn_global_load_async_to_lds_b128(
          (gb128_t*)g, (lb128_t*)l, 0, 0);
#else
      *(float4*)l = *(const float4*)g;
#endif
    }
  };

  // --- Preload A tile into registers in exact 16x4 f32 fragment layout ----
  float a_reg[128];
#pragma unroll
  for (int t = 0; t < 64; ++t) {
    int d = 4 * t + c0;
    a_reg[2 * t]     = xbase[(size_t)d * 1024];
    a_reg[2 * t + 1] = xbase[(size_t)(d + 1) * 1024];
  }

  float minv[8];
  int   mini[8];
#pragma unroll
  for (int j = 0; j < 8; ++j) { minv[j] = 3.4e38f; mini[j] = 0; }

  // Prologue: stage tile-pair 0 into buffer 0.
  stage(0, 0);
  wait_async_zero();
  __syncthreads();

  for (int ktp = 0; ktp < KCODES / 32; ++ktp) {
    const int kbase = ktp * 32;
    const int cur   = ktp & 1;

    // Kick off async staging of the NEXT tile-pair into the other buffer
    // (its last readers finished before the barrier we just passed).
    if (ktp + 1 < KCODES / 32) stage(1 - cur, kbase + 32);

    v8f acc0 = {};
    v8f acc1 = {};
    const int b0base = nn * LDS_W + c0;            // tile columns kbase..+15
    const int b1base = (nn + 16) * LDS_W + c0;     // tile columns kbase+16..+31
#pragma unroll
    for (int t = 0; t < 64; ++t) {
      v2f a  = { a_reg[2 * t], a_reg[2 * t + 1] };
      v2f b0 = *(const v2f*)&eL[cur][b0base + 4 * t];  // conflict-free ds_load_b64
      v2f b1 = *(const v2f*)&eL[cur][b1base + 4 * t];
      acc0 = __builtin_amdgcn_wmma_f32_16x16x4_f32(
          false, a, false, b0, (short)0, acc0, false, false);
      acc1 = __builtin_amdgcn_wmma_f32_16x16x4_f32(
          false, a, false, b1, (short)0, acc1, false, false);
    }

    // Fold scores into running per-lane min (column = kbase+nn / kbase+16+nn).
    const int   k0 = kbase + nn;
    const int   k1 = k0 + 16;
    const float e0 = eTe[k0];
    const float e1 = eTe[k1];
#pragma unroll
    for (int j = 0; j < 8; ++j) {
      float s0 = e0 - 2.0f * acc0[j];     // C layout: VGPR j -> M = j (+8 hi)
      if (s0 < minv[j]) { minv[j] = s0; mini[j] = k0; }
      float s1 = e1 - 2.0f * acc1[j];
      if (s1 < minv[j]) { minv[j] = s1; mini[j] = k1; }
    }

    wait_async_zero();                    // own async copies landed in LDS
    __syncthreads();                      // everyone's copies visible; reads done
  }

  // --- Cross-lane argmin over the 16 column-slots (stays in 16-lane halves).
#pragma unroll
  for (int j = 0; j < 8; ++j) {
    float v  = minv[j];
    int   ki = mini[j];
#pragma unroll
    for (int off = 8; off >= 1; off >>= 1) {
      float ov = __shfl_xor(v, off, 32);
      int   ok = __shfl_xor(ki, off, 32);
      if (ov < v || (ov == v && ok < ki)) { v = ov; ki = ok; }
    }
    if (nn == 0) {
      int n_out = mbase + j + ((lane >> 4) << 3);  // M=j (lane 0), M=j+8 (lane 16)
      idxbuf[n_out]    = ki;
      idx_out_f[n_out] = (float)ki;
    }
  }
}

// ---------------------------------------------------------------------------
// Kernel 3: gather quantized output (coalesced NCHW) + per-block SSE partials.
// ---------------------------------------------------------------------------
__global__ void __launch_bounds__(256)
k_quant(const float* __restrict__ in, const float* __restrict__ emb,
        const int* __restrict__ idxbuf, float* __restrict__ out,
        float* __restrict__ partials) {
  __shared__ float red[256];
  int tid = blockIdx.x * 256 + threadIdx.x;   // 32768 blocks -> QELEMS threads
  int b  = tid >> 18;
  int c  = (tid >> 10) & 255;
  int hw = tid & 1023;
  int n  = (b << 10) | hw;
  int k  = idxbuf[n];
  float q = emb[(size_t)k * DIM + c];         // L2-resident gather
  float x = in[tid];
  out[tid] = q;                               // straight-through value
  float d = q - x;
  red[threadIdx.x] = d * d;
  __syncthreads();
#pragma unroll
  for (int s = 128; s > 0; s >>= 1) {
    if (threadIdx.x < s) red[threadIdx.x] += red[threadIdx.x + s];
    __syncthreads();
  }
  if (threadIdx.x == 0) partials[blockIdx.x] = red[0];
}

// ---------------------------------------------------------------------------
// Kernel 4: deterministic final loss reduction: loss = 1.25 * mean(sse).
// ---------------------------------------------------------------------------
__global__ void k_loss(const float* __restrict__ partials,
                       float* __restrict__ loss_out) {
  __shared__ float red[256];
  float s = 0.f;
  for (int i = threadIdx.x; i < 32768; i += 256) s += partials[i];
  red[threadIdx.x] = s;
  __syncthreads();
#pragma unroll
  for (int st = 128; st > 0; st >>= 1) {
    if (threadIdx.x < st) red[threadIdx.x] += red[threadIdx.x + st];
    __syncthreads();
  }
  if (threadIdx.x == 0) loss_out[0] = red[0] * (1.25f / (float)QELEMS);
}

// ---------------------------------------------------------------------------
extern "C" void kernel_launch(void* const* d_in, const int* in_sizes, int n_in,
                              void* d_out, int out_size, void* d_ws, size_t ws_size,
                              hipStream_t stream) {
  const float* inputs = (const float*)d_in[0];
  const float* emb    = (const float*)d_in[1];
  float* out = (float*)d_out;

  // Workspace layout (bytes): eTe[1024]f @0, idx[32768]i @4096, partials[32768]f @135168
  float* ws_eTe      = (float*)d_ws;
  int*   ws_idx      = (int*)((char*)d_ws + 4096);
  float* ws_partials = (float*)((char*)d_ws + 4096 + 32768 * sizeof(int));

  float* out_quant = out;                 // [QELEMS]
  float* out_loss  = out + QELEMS;        // [1]
  float* out_idx   = out + QELEMS + 1;    // [NROWS]

  k_ete<<<KCODES / 256, 256, 0, stream>>>(emb, ws_eTe);
  k_argmin<<<NROWS / 128, 256, 0, stream>>>(inputs, emb, ws_eTe, ws_idx, out_idx);
  k_quant<<<QELEMS / 256, 256, 0, stream>>>(inputs, emb, ws_idx, out_quant, ws_partials);
  k_loss<<<1, 256, 0, stream>>>(ws_partials, out_loss);
}